// spec_select_84009560310230
// MI455X (gfx1250) — compile-verified
//
#include <hip/hip_runtime.h>

typedef __attribute__((ext_vector_type(2))) float v2f;
typedef __attribute__((ext_vector_type(8))) float v8f;

__device__ __forceinline__ float sigmoidf_(float x) { return 1.0f / (1.0f + __expf(-x)); }

// ---------------------------------------------------------------------------
// Kernel A: global average pool of R and T -> gapR/gapT [8,256]
// grid 4096 (2 tensors * 8*256 planes), 256 threads each reduce 4096 elems
// ---------------------------------------------------------------------------
__global__ void gap_kernel(const float* __restrict__ R, const float* __restrict__ T,
                           float* __restrict__ gapR, float* __restrict__ gapT) {
  __shared__ float red[256];
  int b = blockIdx.x;
  const float* X = (b & 2048) ? T : R;
  float* out = (b & 2048) ? gapT : gapR;
  int idx = b & 2047;  // n*256 + c
  const float* p = X + ((size_t)idx << 12);
  float s = 0.f;
  for (int i = threadIdx.x; i < 4096; i += 256) s += p[i];
  red[threadIdx.x] = s;
  __syncthreads();
  for (int off = 128; off; off >>= 1) {
    if (threadIdx.x < off) red[threadIdx.x] += red[threadIdx.x + off];
    __syncthreads();
  }
  if (threadIdx.x == 0) out[idx] = red[0] * (1.0f / 4096.0f);
}

// ---------------------------------------------------------------------------
// Kernel B: all tiny MLPs. 1 block, 256 threads.
//  - involution filter logits (72) + BN + per-group softmax -> filtR/filtT [8,8,9]
//  - channel attention gates wR1/wT1 (from gapR via ff1), wR2/wT2 (gapT via ff2)
// ---------------------------------------------------------------------------
__global__ __launch_bounds__(256) void head_kernel(
    const float* __restrict__ gapR, const float* __restrict__ gapT,
    const float* __restrict__ conv_w, const float* __restrict__ bng,
    const float* __restrict__ bnb, const float* __restrict__ bnm, const float* __restrict__ bnv,
    const float* __restrict__ ff1W1, const float* __restrict__ ff1b1,
    const float* __restrict__ ff1W2R, const float* __restrict__ ff1b2R,
    const float* __restrict__ ff1W2T, const float* __restrict__ ff1b2T,
    const float* __restrict__ ff2W1, const float* __restrict__ ff2b1,
    const float* __restrict__ ff2W2R, const float* __restrict__ ff2b2R,
    const float* __restrict__ ff2W2T, const float* __restrict__ ff2b2T,
    float* __restrict__ filtR, float* __restrict__ filtT,
    float* __restrict__ wR1, float* __restrict__ wT1,
    float* __restrict__ wR2, float* __restrict__ wT2) {
  __shared__ float g1[256], g2[256], s1[256], s2[256], lf1[72], lf2[72];
  int t = threadIdx.x;
  for (int n = 0; n < 8; ++n) {
    g1[t] = gapR[n * 256 + t];
    g2[t] = gapT[n * 256 + t];
    __syncthreads();
    if (t < 72) {
      float a = 0.f, b = 0.f;
      for (int c = 0; c < 256; ++c) {
        float w = conv_w[t * 256 + c];
        a += g1[c] * w;
        b += g2[c] * w;
      }
      float inv = rsqrtf(bnv[t] + 1e-5f);
      lf1[t] = (a - bnm[t]) * inv * bng[t] + bnb[t];
      lf2[t] = (b - bnm[t]) * inv * bng[t] + bnb[t];
    }
    {
      float a = ff1b1[t], b = ff2b1[t];
      for (int c = 0; c < 256; ++c) {
        a += g1[c] * ff1W1[c * 256 + t];
        b += g2[c] * ff2W1[c * 256 + t];
      }
      s1[t] = fmaxf(a, 0.f);
      s2[t] = fmaxf(b, 0.f);
    }
    __syncthreads();
    if (t < 16) {
      const float* src = (t < 8) ? lf1 : lf2;
      float* dst = (t < 8) ? filtR : filtT;
      int g = t & 7;
      float mx = -1e30f;
      for (int k = 0; k < 9; ++k) mx = fmaxf(mx, src[g * 9 + k]);
      float e[9], sm = 0.f;
      for (int k = 0; k < 9; ++k) { e[k] = __expf(src[g * 9 + k] - mx); sm += e[k]; }
      float is = 1.f / sm;
      for (int k = 0; k < 9; ++k) dst[n * 72 + g * 9 + k] = e[k] * is;
    }
    {
      float a = ff1b2R[t], b = ff1b2T[t], c2 = ff2b2R[t], d = ff2b2T[t];
      for (int h = 0; h < 256; ++h) {
        float x1 = s1[h], x2 = s2[h];
        a  += x1 * ff1W2R[h * 256 + t];
        b  += x1 * ff1W2T[h * 256 + t];
        c2 += x2 * ff2W2R[h * 256 + t];
        d  += x2 * ff2W2T[h * 256 + t];
      }
      wR1[n * 256 + t] = sigmoidf_(a);
      wT1[n * 256 + t] = sigmoidf_(b);
      wR2[n * 256 + t] = sigmoidf_(c2);
      wT2[n * 256 + t] = sigmoidf_(d);
    }
    __syncthreads();
  }
}

// ---------------------------------------------------------------------------
// Kernel C: fused involution + gated recombine + 256->16 1x1 conv via WMMA.
// grid = N*H = 512 blocks (one image row), block = 128 threads (4 waves),
// each wave owns 16 pixels; K=256 channels in 64 chunks of 4 for f32 WMMA.
// A layout (16x4 f32): lanes 0-15 hold {K0,K1}, lanes 16-31 hold {K2,K3}.
// B layout (4x16 f32): mirror, N striped over 16 lanes per half.
// D (16x16 f32): 8 VGPRs, lanes 0-15 -> M=r, lanes 16-31 -> M=r+8.
// ---------------------------------------------------------------------------
__global__ __launch_bounds__(128) void mid_kernel(
    const float* __restrict__ R, const float* __restrict__ T,
    const float* __restrict__ filtR, const float* __restrict__ filtT,
    const float* __restrict__ wR1, const float* __restrict__ wT1,
    const float* __restrict__ wR2, const float* __restrict__ wT2,
    const float* __restrict__ fv_wR, const float* __restrict__ fv_bR,
    const float* __restrict__ fv_wT, const float* __restrict__ fv_bT,
    float* __restrict__ Rf, float* __restrict__ Tf) {
  int n = blockIdx.x >> 6;
  int y = blockIdx.x & 63;
  int wave = threadIdx.x >> 5;
  int lane = threadIdx.x & 31;
  int half = lane >> 4;
  int l16 = lane & 15;
  int px = wave * 16 + l16;

  // reflect-pad neighbor indices
  int ym = (y > 0) ? y - 1 : 1;
  int yp = (y < 63) ? y + 1 : 62;
  int xm = (px > 0) ? px - 1 : 1;
  int xp = (px < 63) ? px + 1 : 62;
  int ro_m = ym * 64, ro_0 = y * 64, ro_p = yp * 64;

  const float* fR = filtR + n * 72;
  const float* fT = filtT + n * 72;
  const float* wlR = wR1 + n * 256;
  const float* whR = wT1 + n * 256;
  const float* wlT = wR2 + n * 256;
  const float* whT = wT2 + n * 256;
  const size_t baseNC = (size_t)n * 256 * 4096;

  v8f accR = {};
  v8f accT = {};
  for (int kc = 0; kc < 64; ++kc) {
    int kA = kc * 4 + 2 * half;
    v2f aR, aT, bR, bT;
    aR.x = fv_wR[l16 * 256 + kA];     aR.y = fv_wR[l16 * 256 + kA + 1];
    aT.x = fv_wT[l16 * 256 + kA];     aT.y = fv_wT[l16 * 256 + kA + 1];
#pragma unroll
    for (int j = 0; j < 2; ++j) {
      int c = kA + j;
      const float* Xc = R + baseNC + (size_t)c * 4096;
      const float* Yc = T + baseNC + (size_t)c * 4096;
      const float* fg = fR + (c >> 5) * 9;
      const float* fh = fT + (c >> 5) * 9;
      float lowR = fg[0] * Xc[ro_m + xm] + fg[1] * Xc[ro_m + px] + fg[2] * Xc[ro_m + xp]
                 + fg[3] * Xc[ro_0 + xm] + fg[4] * Xc[ro_0 + px] + fg[5] * Xc[ro_0 + xp]
                 + fg[6] * Xc[ro_p + xm] + fg[7] * Xc[ro_p + px] + fg[8] * Xc[ro_p + xp];
      float lowT = fh[0] * Yc[ro_m + xm] + fh[1] * Yc[ro_m + px] + fh[2] * Yc[ro_m + xp]
                 + fh[3] * Yc[ro_0 + xm] + fh[4] * Yc[ro_0 + px] + fh[5] * Yc[ro_0 + xp]
                 + fh[6] * Yc[ro_p + xm] + fh[7] * Yc[ro_p + px] + fh[8] * Yc[ro_p + xp];
      float cenR = Xc[ro_0 + px];
      float cenT = Yc[ro_0 + px];
      float vR = wlR[c] * lowR + whR[c] * (cenR - lowR);
      float vT = wlT[c] * lowT + whT[c] * (cenT - lowT);
      if (j == 0) { bR.x = vR; bT.x = vT; } else { bR.y = vR; bT.y = vT; }
    }
    accR = __builtin_amdgcn_wmma_f32_16x16x4_f32(false, aR, false, bR, (short)0, accR, false, false);
    accT = __builtin_amdgcn_wmma_f32_16x16x4_f32(false, aT, false, bT, (short)0, accT, false, false);
  }
  size_t rowoff = (size_t)y * 64 + px;
#pragma unroll
  for (int r = 0; r < 8; ++r) {
    int hc = r + 8 * half;
    size_t o = ((size_t)(n * 16 + hc)) * 4096 + rowoff;
    Rf[o] = accR[r] + fv_bR[hc];
    Tf[o] = accT[r] + fv_bT[hc];
  }
}

// ---------------------------------------------------------------------------
// Kernel D: GAP of Gf = Rf+Tf -> gapG [8,16]. 128 blocks.
// ---------------------------------------------------------------------------
__global__ void gapg_kernel(const float* __restrict__ Rf, const float* __restrict__ Tf,
                            float* __restrict__ gapG) {
  __shared__ float red[256];
  int idx = blockIdx.x;  // n*16 + hc
  const float* a = Rf + ((size_t)idx << 12);
  const float* b = Tf + ((size_t)idx << 12);
  float s = 0.f;
  for (int i = threadIdx.x; i < 4096; i += 256) s += a[i] + b[i];
  red[threadIdx.x] = s;
  __syncthreads();
  for (int off = 128; off; off >>= 1) {
    if (threadIdx.x < off) red[threadIdx.x] += red[threadIdx.x + off];
    __syncthreads();
  }
  if (threadIdx.x == 0) gapG[idx] = red[0] * (1.0f / 4096.0f);
}

// ---------------------------------------------------------------------------
// Kernel E: wG = sigmoid(gapG @ fv_Lw + fv_Lb)  [8,16]. 1 block.
// ---------------------------------------------------------------------------
__global__ void wg_kernel(const float* __restrict__ gapG, const float* __restrict__ Lw,
                          const float* __restrict__ Lb, float* __restrict__ wG) {
  int t = threadIdx.x;
  if (t >= 128) return;
  int n = t >> 4, hc = t & 15;
  float a = Lb[hc];
  for (int j = 0; j < 16; ++j) a += gapG[n * 16 + j] * Lw[j * 16 + hc];
  wG[t] = sigmoidf_(a);
}

// ---------------------------------------------------------------------------
// Kernel F: fovea softmax over W per (hc,row) + global gate, then 16->256
// 1x1 conv via WMMA. grid = N*H = 512, 256 threads (8 waves).
// ---------------------------------------------------------------------------
__global__ __launch_bounds__(256) void out_kernel(
    const float* __restrict__ Rf, const float* __restrict__ Tf,
    const float* __restrict__ wG, const float* __restrict__ sRp, const float* __restrict__ sTp,
    const float* __restrict__ wO, const float* __restrict__ bO, float* __restrict__ out) {
  __shared__ float sA[16 * 64];  // Rf row, then reused as fused[hc][x]
  __shared__ float sB[16 * 64];  // Tf row
  int n = blockIdx.x >> 6, y = blockIdx.x & 63;
  int tid = threadIdx.x;
  for (int i = tid; i < 1024; i += 256) {
    int hc = i >> 6, x = i & 63;
    size_t o = ((size_t)(n * 16 + hc)) * 4096 + (size_t)y * 64 + x;
    sA[i] = Rf[o];
    sB[i] = Tf[o];
  }
  __syncthreads();
  if (tid < 16) {
    float sR = sRp[0], sT = sTp[0], wg = wG[n * 16 + tid];
    float mR = -1e30f, mT = -1e30f;
    for (int x = 0; x < 64; ++x) {
      mR = fmaxf(mR, sR * sA[tid * 64 + x]);
      mT = fmaxf(mT, sT * sB[tid * 64 + x]);
    }
    float zR = 0.f, zT = 0.f;
    for (int x = 0; x < 64; ++x) {
      zR += __expf(sR * sA[tid * 64 + x] - mR);
      zT += __expf(sT * sB[tid * 64 + x] - mT);
    }
    float iR = 1.f / zR, iT = 1.f / zT;
    for (int x = 0; x < 64; ++x) {
      float r = sA[tid * 64 + x], t2 = sB[tid * 64 + x];
      sA[tid * 64 + x] = __expf(sR * r - mR) * iR * r + __expf(sT * t2 - mT) * iT * t2 + (r + t2) * wg;
    }
  }
  __syncthreads();
  int wave = tid >> 5, lane = tid & 31, half = lane >> 4, l16 = lane & 15;
  for (int t = wave; t < 64; t += 8) {  // 16 oc-tiles x 4 x-tiles
    int ocb = (t >> 2) * 16, xb = (t & 3) * 16;
    v8f acc = {};
#pragma unroll
    for (int k0 = 0; k0 < 16; k0 += 4) {
      int k = k0 + 2 * half;
      v2f a, b;
      a.x = wO[(ocb + l16) * 16 + k];
      a.y = wO[(ocb + l16) * 16 + k + 1];
      b.x = sA[k * 64 + xb + l16];
      b.y = sA[(k + 1) * 64 + xb + l16];
      acc = __builtin_amdgcn_wmma_f32_16x16x4_f32(false, a, false, b, (short)0, acc, false, false);
    }
#pragma unroll
    for (int r = 0; r < 8; ++r) {
      int oc = ocb + r + 8 * half;
      out[((size_t)(n * 256 + oc)) * 4096 + (size_t)y * 64 + xb + l16] = acc[r] + bO[oc];
    }
  }
}

// ---------------------------------------------------------------------------
extern "C" void kernel_launch(void* const* d_in, const int* in_sizes, int n_in,
                              void* d_out, int out_size, void* d_ws, size_t ws_size,
                              hipStream_t stream) {
  const float* R = (const float*)d_in[0];
  const float* T = (const float*)d_in[1];
  const float* conv_w = (const float*)d_in[2];
  const float* bng = (const float*)d_in[3];
  const float* bnb = (const float*)d_in[4];
  const float* bnm = (const float*)d_in[5];
  const float* bnv = (const float*)d_in[6];
  const float* ff1W1 = (const float*)d_in[7];
  const float* ff1b1 = (const float*)d_in[8];
  const float* ff1W2R = (const float*)d_in[9];
  const float* ff1b2R = (const float*)d_in[10];
  const float* ff1W2T = (const float*)d_in[11];
  const float* ff1b2T = (const float*)d_in[12];
  const float* ff2W1 = (const float*)d_in[13];
  const float* ff2b1 = (const float*)d_in[14];
  const float* ff2W2R = (const float*)d_in[15];
  const float* ff2b2R = (const float*)d_in[16];
  const float* ff2W2T = (const float*)d_in[17];
  const float* ff2b2T = (const float*)d_in[18];
  const float* fv_wR = (const float*)d_in[19];
  const float* fv_bR = (const float*)d_in[20];
  const float* fv_wT = (const float*)d_in[21];
  const float* fv_bT = (const float*)d_in[22];
  const float* fv_Lw = (const float*)d_in[23];
  const float* fv_Lb = (const float*)d_in[24];
  const float* fv_wO = (const float*)d_in[25];
  const float* fv_bO = (const float*)d_in[26];
  const float* fv_sR = (const float*)d_in[27];
  const float* fv_sT = (const float*)d_in[28];

  float* ws = (float*)d_ws;
  float* gapR = ws;               // 2048
  float* gapT = gapR + 2048;      // 2048
  float* filtR = gapT + 2048;     // 576
  float* filtT = filtR + 576;     // 576
  float* wR1 = filtT + 576;       // 2048
  float* wT1 = wR1 + 2048;        // 2048
  float* wR2 = wT1 + 2048;        // 2048
  float* wT2 = wR2 + 2048;        // 2048
  float* Rf = wT2 + 2048;         // 524288
  float* Tf = Rf + 524288;        // 524288
  float* gapG = Tf + 524288;      // 128
  float* wG = gapG + 128;         // 128

  // first output = R, passed through
  hipMemcpyAsync(d_out, d_in[0], (size_t)8 * 256 * 64 * 64 * sizeof(float),
                 hipMemcpyDeviceToDevice, stream);

  gap_kernel<<<4096, 256, 0, stream>>>(R, T, gapR, gapT);
  head_kernel<<<1, 256, 0, stream>>>(gapR, gapT, conv_w, bng, bnb, bnm, bnv,
                                     ff1W1, ff1b1, ff1W2R, ff1b2R, ff1W2T, ff1b2T,
                                     ff2W1, ff2b1, ff2W2R, ff2b2R, ff2W2T, ff2b2T,
                                     filtR, filtT, wR1, wT1, wR2, wT2);
  mid_kernel<<<512, 128, 0, stream>>>(R, T, filtR, filtT, wR1, wT1, wR2, wT2,
                                      fv_wR, fv_bR, fv_wT, fv_bT, Rf, Tf);
  gapg_kernel<<<128, 256, 0, stream>>>(Rf, Tf, gapG);
  wg_kernel<<<1, 128, 0, stream>>>(gapG, fv_Lw, fv_Lb, wG);
  out_kernel<<<512, 256, 0, stream>>>(Rf, Tf, wG, fv_sR, fv_sT, fv_wO, fv_bO,
                                      (float*)d_out + (size_t)8 * 256 * 64 * 64);
}